// IDWT3D_19155554140416
// MI455X (gfx1250) — compile-verified
//
#include <hip/hip_runtime.h>

// ---------------------------------------------------------------------------
// IDWT3D (inverse 3-D Haar) for MI455X / gfx1250.
// HBM-bound: ~453 MB traffic -> ~19us floor @ 23.3 TB/s; only ~0.9 GFLOP.
// Strategy: one wave32 <-> one 48-voxel W-row; 3 tiles of 16 voxels, each a
// pair of chained V_WMMA_F32_16X16X4_F32 (K = 8 subbands). All row index math
// is wave-uniform (readfirstlane -> SALU); per-tile loads/stores use constant
// immediate offsets so VALU stays far below the issue roofline.
// ---------------------------------------------------------------------------

typedef float v2f __attribute__((ext_vector_type(2)));
typedef float v8f __attribute__((ext_vector_type(8)));

constexpr int B_ = 2, C_ = 32, D_ = 48, H_ = 48, W_ = 48;
constexpr int HW  = H_ * W_;        // 2304
constexpr int SP  = D_ * HW;        // 110592 voxels per (b,c)
constexpr int OUT_BC = 8 * SP;      // output floats per (b,c)  (2D*2H*2W)
constexpr int OUT_D  = 4 * HW;      // output depth-slice stride (2H*2W)
constexpr int OUT_H  = 2 * W_;      // output row stride (2W)

__global__ __launch_bounds__(256) void idwt3d_wmma_kernel(
    const float* __restrict__ low,    // [B,C,D,H,W]
    const float* __restrict__ highs,  // [B,C,7,D,H,W]
    const float* __restrict__ kern,   // [8,2,2,2] flat: s*8 + i*4 + j*2 + k
    float* __restrict__ out)          // [B,C,2D,2H,2W]
{
  const int lane = threadIdx.x & 31;
  const int hl   = lane >> 4;        // lane half selects K-pair (ISA 7.12.2)
  const int l16  = lane & 15;

  // ---- wave-uniform row decomposition, forced onto the scalar path --------
  const int wv  = __builtin_amdgcn_readfirstlane((int)(threadIdx.x >> 5));
  const int bc  = blockIdx.y;                 // (b,c) pair: 0..63
  const int row = blockIdx.x * 8 + wv;        // d*H + h within (b,c): 0..2303
  const int d   = row / H_;
  const int h   = row - d * H_;
  const int sp0 = d * HW + h * W_;            // row start within the volume

  const float* lowB = low   + (long)bc * SP + sp0;
  const float* hiB  = highs + (long)bc * 7 * SP + sp0;

  // ---- per-lane A-operand plane pointers (computed once) ------------------
  // f32 A (16x4) layout: lanes 0-15 carry K=0,1 / lanes 16-31 carry K=2,3,
  // so hl=0 supplies subbands {0,1,4,5}, hl=1 supplies {2,3,6,7}.
  const float* p00 = (hl ? hiB + 1 * SP : lowB) + l16;  // K-chunk0, vgpr0
  const float* p01 = hiB + (hl ? 2 : 0) * SP + l16;     // K-chunk0, vgpr1
  const float* p10 = hiB + (hl ? 5 : 3) * SP + l16;     // K-chunk1, vgpr0
  const float* p11 = hiB + (hl ? 6 : 4) * SP + l16;     // K-chunk1, vgpr1

  // ---- B operand: 4x16 Haar matrix (subband-row x position), cols 8..15 = 0
  const int  s0 = 2 * hl;
  const bool bp = (l16 < 8);
  v2f b01, b23;
  b01.x = bp ? kern[(s0    ) * 8 + l16] : 0.0f;
  b01.y = bp ? kern[(s0 + 1) * 8 + l16] : 0.0f;
  b23.x = bp ? kern[(s0 + 4) * 8 + l16] : 0.0f;
  b23.y = bp ? kern[(s0 + 5) * 8 + l16] : 0.0f;

  // ---- per-lane store base (computed once): position p = l16, and the
  // accumulator row r maps to voxel w = 16t + r + 8*hl  ->  out elem
  // rowbase + k + 2w  =  base + (32t + 2r) with 16*hl folded into base.
  const int p  = l16 & 7;
  const int ii = p >> 2, jj = (p >> 1) & 1, kk = p & 1;
  float* outB = out + (long)bc * OUT_BC
              + (2 * d + ii) * OUT_D + (2 * h + jj) * OUT_H + kk + 16 * hl;

  #pragma unroll
  for (int t = 0; t < 3; ++t) {               // 3 x 16 voxels = full row
    v2f a01, a23;
    a01.x = p00[16 * t];  a01.y = p01[16 * t];
    a23.x = p10[16 * t];  a23.y = p11[16 * t];

    // D = A0*B0 + A1*B1 (EXEC all-1s here; store branch below reconverges)
    v8f acc = {};
    acc = __builtin_amdgcn_wmma_f32_16x16x4_f32(
        false, a01, false, b01, (short)0, acc, false, false);
    acc = __builtin_amdgcn_wmma_f32_16x16x4_f32(
        false, a23, false, b23, (short)0, acc, false, false);

    if (l16 < 8) {
      #pragma unroll
      for (int r = 0; r < 8; ++r)
        outB[32 * t + 2 * r] = acc[r];        // immediate-offset stores
    }
  }
}

extern "C" void kernel_launch(void* const* d_in, const int* in_sizes, int n_in,
                              void* d_out, int out_size, void* d_ws, size_t ws_size,
                              hipStream_t stream) {
  const float* low   = (const float*)d_in[0];   // [2,32,48,48,48]
  const float* highs = (const float*)d_in[1];   // [2,32,7,48,48,48]
  const float* kern  = (const float*)d_in[2];   // [8,2,2,2]
  float*       out   = (float*)d_out;           // [2,32,96,96,96]

  // one wave per W-row: D*H = 2304 rows per (b,c), 8 waves per block
  dim3 grid(D_ * H_ / 8, B_ * C_);              // (288, 64)
  idwt3d_wmma_kernel<<<grid, 256, 0, stream>>>(low, highs, kern, out);
}